// CRF_26671746908199
// MI455X (gfx1250) — compile-verified
//
#include <hip/hip_runtime.h>

// CRF Viterbi decode for MI455X (gfx1250, wave32).
// Stage 1: emissions GEMM (32768x512 @ 512x22 fp32) via V_WMMA_F32_16X16X4_F32,
//          with async global->LDS staging (ASYNCcnt path) + LDS fragment reads.
// Stage 2: Viterbi forward scan, one wave32 per batch, lane = class.
// Stage 3: backtrace, one thread per batch.

#define IN_DIM    512
#define NUM_C     22          // NUM_CLASSES + 2
#define START_C   20
#define STOP_C    21
#define BATCH     64
#define SEQL      512
#define NEGV      (-100000.0f)

#define KC        128                 // K-chunk staged in LDS
#define NCHUNK    (IN_DIM / KC)       // 4
#define WAVES_PB  4                   // waves per block
#define ROWS_PB   (WAVES_PB * 16)     // 64 output rows per block
#define ASTRIDE   (KC + 4)            // 132: row stride ≡ 4 (mod 64 banks) -> conflict-free
#define WSTRIDE   (IN_DIM + 4)        // 516: same trick for W

typedef __attribute__((ext_vector_type(2))) float v2f;
typedef __attribute__((ext_vector_type(4))) float v4f;
typedef __attribute__((ext_vector_type(8))) float v8f;
typedef __attribute__((ext_vector_type(4))) int   v4i;

typedef __attribute__((address_space(1))) v4i g_v4i;   // global (AS1)
typedef __attribute__((address_space(3))) v4i l_v4i;   // LDS (AS3)

#if defined(__has_builtin)
#if __has_builtin(__builtin_amdgcn_global_load_async_to_lds_b128)
#define HAVE_ASYNC_LDS 1
#endif
#endif

__device__ __forceinline__ void async_wait0() {
    asm volatile("s_wait_asynccnt 0" ::: "memory");
}
__device__ __forceinline__ void ds_wait0() {
    asm volatile("s_wait_dscnt 0" ::: "memory");
}

// 16-byte global -> LDS copy; async (ASYNCcnt) when the builtin exists.
__device__ __forceinline__ void copy16_g2l(const float* g, float* l) {
#ifdef HAVE_ASYNC_LDS
    __builtin_amdgcn_global_load_async_to_lds_b128(
        (g_v4i*)g, (l_v4i*)l, 0, 0);
#else
    *(v4f*)l = *(const v4f*)g;
#endif
}

// ---------------------------------------------------------------------------
// Kernel 1: emissions GEMM. Block = 4 waves = 64 rows; each wave owns a
// 16-row x 32-col tile (cols 22..31 computed, never stored).
// A fragment (16x4 f32): lanes 0-15 rows M=0..15 K={k,k+1}; lanes 16-31 K={k+2,k+3}.
// B fragment (4x16 f32): mirrored, N on lanes. C/D: VGPR v -> row v+8*(lane>=16).
// ---------------------------------------------------------------------------
__global__ __launch_bounds__(128) void crf_emit_gemm(
    const float* __restrict__ x,     // [B*L, IN_DIM]
    const float* __restrict__ W,     // [NUM_C, IN_DIM]
    const float* __restrict__ bias,  // [NUM_C]
    float* __restrict__ emit)        // [B*L, NUM_C]
{
    __shared__ float sW[NUM_C * WSTRIDE];                 // 45,408 B
    __shared__ float sA[2 * WAVES_PB * 16 * ASTRIDE];     // 67,584 B

    const int  tid   = threadIdx.x;
    const int  lane  = tid & 31;
    const int  wv    = tid >> 5;
    const long mBase = (long)blockIdx.x * ROWS_PB + (long)wv * 16;

    // ---- stage W once (22 rows x 512 floats), block-cooperative, coalesced
    {
        const int nf4 = NUM_C * (IN_DIM / 4);             // 2816 float4s
        for (int f = tid; f < nf4; f += 128) {
            const int row = f >> 7;                       // 128 f4 per row
            const int c4  = f & 127;
            copy16_g2l(W + row * IN_DIM + c4 * 4, &sW[row * WSTRIDE + c4 * 4]);
        }
    }

    // ---- per-wave A chunk loader: 16 rows x KC floats, 512B contiguous per row
    auto loadA = [&](int buf, int chunk) {
        float* dst = &sA[(buf * WAVES_PB + wv) * 16 * ASTRIDE];
        const float* src = x + mBase * IN_DIM + chunk * KC;
        #pragma unroll
        for (int r = 0; r < 16; ++r)   // 32 lanes x 16B = one full 512B row
            copy16_g2l(src + r * IN_DIM + lane * 4, dst + r * ASTRIDE + lane * 4);
    };

    loadA(0, 0);
    async_wait0();
    __syncthreads();                  // W visible to all waves

    const int mRow  = lane & 15;
    const int kHalf = (lane >> 4) << 1;            // 0 or 2
    const int n0 = lane & 15;
    const int n1 = 16 + n0;
    const int c0 = n0;                              // < 16 < NUM_C
    const int c1 = (n1 < NUM_C) ? n1 : (NUM_C - 1); // clamp (broadcast in LDS)

    const float* __restrict__ wr0 = &sW[c0 * WSTRIDE + kHalf];
    const float* __restrict__ wr1 = &sW[c1 * WSTRIDE + kHalf];

    v8f acc0 = {};
    v8f acc1 = {};

    for (int chunk = 0; chunk < NCHUNK; ++chunk) {
        const int cur = chunk & 1;
        if (chunk + 1 < NCHUNK) {
            ds_wait0();               // prior DS reads of the other buffer retired
            loadA(cur ^ 1, chunk + 1);
        }
        const float* __restrict__ ar =
            &sA[(cur * WAVES_PB + wv) * 16 * ASTRIDE + mRow * ASTRIDE + kHalf];
        const float* __restrict__ w0c = wr0 + chunk * KC;
        const float* __restrict__ w1c = wr1 + chunk * KC;

        #pragma unroll 4
        for (int k = 0; k < KC; k += 4) {
            const v2f a  = *(const v2f*)(ar  + k);
            const v2f b0 = *(const v2f*)(w0c + k);
            const v2f b1 = *(const v2f*)(w1c + k);
            acc0 = __builtin_amdgcn_wmma_f32_16x16x4_f32(
                false, a, false, b0, (short)0, acc0, false, false);
            acc1 = __builtin_amdgcn_wmma_f32_16x16x4_f32(
                false, a, false, b1, (short)0, acc1, false, false);
        }
        if (chunk + 1 < NCHUNK) async_wait0();   // next chunk resident
    }

    const int rowOff = (lane >> 4) << 3;   // 0 or 8
    const float bb0 = bias[c0];
    const float bb1 = bias[c1];
    #pragma unroll
    for (int v = 0; v < 8; ++v) {
        const long m = mBase + rowOff + v;
        emit[m * NUM_C + n0] = acc0[v] + bb0;
        if (n1 < NUM_C) emit[m * NUM_C + n1] = acc1[v] + bb1;
    }
}

// ---------------------------------------------------------------------------
// Kernel 2: Viterbi forward scan. One wave32 per batch; lane = "to" class.
// ---------------------------------------------------------------------------
__global__ __launch_bounds__(32) void crf_viterbi_fwd(
    const float* __restrict__ emit,      // [B*L, NUM_C]
    const float* __restrict__ mask,      // [B, L]
    const float* __restrict__ trans,     // [NUM_C, NUM_C]
    float* __restrict__ maxScoreOut,     // [B, NUM_C]
    unsigned char* __restrict__ bp,      // [L, B, NUM_C]
    int* __restrict__ bestTags)          // [B]
{
    const int b  = blockIdx.x;
    const int to = threadIdx.x;
    const bool valid = (to < NUM_C);

    float trow[NUM_C];
    #pragma unroll
    for (int f = 0; f < NUM_C; ++f)
        trow[f] = valid ? trans[to * NUM_C + f] : NEGV;

    float score = (to == START_C) ? 0.0f : NEGV;

    for (int t = 0; t < SEQL; ++t) {
        float best = -3.0e38f;
        int   bi   = 0;
        #pragma unroll
        for (int f = 0; f < NUM_C; ++f) {
            const float v = __shfl(score, f, 32) + trow[f];
            if (v > best) { best = v; bi = f; }   // first max wins (argmax)
        }
        if (valid)
            bp[((long)t * BATCH + b) * NUM_C + to] = (unsigned char)bi;

        const float m = mask[b * SEQL + t];
        if (valid) {
            const float e  = emit[((long)b * SEQL + t) * NUM_C + to];
            const float st = best + e;
            score = (m > 0.0f) ? st : score;
        }
    }

    score = valid ? (score + trans[STOP_C * NUM_C + to]) : -3.0e38f;
    if (valid) maxScoreOut[b * NUM_C + to] = score;

    float best = -3.0e38f;
    int   bt   = 0;
    #pragma unroll
    for (int f = 0; f < NUM_C; ++f) {
        const float v = __shfl(score, f, 32);
        if (v > best) { best = v; bt = f; }
    }
    if (threadIdx.x == 0) bestTags[b] = bt;
}

// ---------------------------------------------------------------------------
// Kernel 3: backtrace. One thread per batch.
// ---------------------------------------------------------------------------
__global__ __launch_bounds__(64) void crf_backtrace(
    const float* __restrict__ mask,
    const unsigned char* __restrict__ bp,
    const int* __restrict__ bestTags,
    int* __restrict__ path)              // [B, L]
{
    const int b = threadIdx.x;
    float sl = 0.0f;
    for (int t = 0; t < SEQL; ++t) sl += mask[b * SEQL + t];
    const int seqLen = (int)sl;

    int tag = bestTags[b];
    for (int t = SEQL - 1; t >= 0; --t) {
        const bool active = (t < seqLen);
        path[b * SEQL + t] = active ? tag : 0;
        const int prev = (int)bp[((long)t * BATCH + b) * NUM_C + tag];
        tag = active ? prev : tag;
    }
}

// ---------------------------------------------------------------------------
extern "C" void kernel_launch(void* const* d_in, const int* in_sizes, int n_in,
                              void* d_out, int out_size, void* d_ws, size_t ws_size,
                              hipStream_t stream)
{
    const float* x     = (const float*)d_in[0];  // [B, L, IN_DIM]
    const float* mask  = (const float*)d_in[1];  // [B, L]
    const float* W     = (const float*)d_in[2];  // [C, IN_DIM]
    const float* bias  = (const float*)d_in[3];  // [C]
    const float* trans = (const float*)d_in[4];  // [C, C]

    char* ws = (char*)d_ws;
    const size_t emitBytes = (size_t)BATCH * SEQL * NUM_C * sizeof(float); // 2,883,584
    const size_t bpBytes   = (size_t)SEQL * BATCH * NUM_C;                 //   720,896
    float*         emit     = (float*)ws;
    unsigned char* bp       = (unsigned char*)(ws + emitBytes);
    int*           bestTags = (int*)(ws + emitBytes + bpBytes);            // 4B aligned

    float* maxOut  = (float*)d_out;                         // [B, C]
    int*   pathOut = (int*)((float*)d_out + BATCH * NUM_C); // [B, L] int32

    // Stage 1: 32768 rows / 64 per block = 512 blocks of 128 threads (4 waves).
    crf_emit_gemm<<<dim3(512), dim3(128), 0, stream>>>(x, W, bias, emit);
    // Stage 2: one wave per batch.
    crf_viterbi_fwd<<<dim3(BATCH), dim3(32), 0, stream>>>(emit, mask, trans,
                                                          maxOut, bp, bestTags);
    // Stage 3: one thread per batch.
    crf_backtrace<<<dim3(1), dim3(64), 0, stream>>>(mask, bp, bestTags, pathOut);
}